// VQVAE_63282048139989
// MI455X (gfx1250) — compile-verified
//
#include <hip/hip_runtime.h>
#include <cstdint>

// VQ-VAE nearest-codebook lookup for MI455X (gfx1250, wave32, WMMA).
// Output-BW bound: one-hot output is 536 MB (~25 us floor @ 23.3 TB/s),
// so compute in exact fp32 via V_WMMA_F32_16X16X4_F32.
// B tiles are double-buffered in LDS via GLOBAL_LOAD_ASYNC_TO_LDS_B128
// (ASYNCcnt path), B fragments preloaded to VGPRs, dual accumulator chains.

typedef __attribute__((ext_vector_type(2))) float v2f;
typedef __attribute__((ext_vector_type(8))) float v8f;

#define BATCH      8192
#define NUM_CODES  16
#define BOOK_SIZE  1024
#define EMBED_DIM  64

#define LDS_STRIDE 68                 // pad 64 -> 68 floats: spread LDS banks
#define NTILES     (BOOK_SIZE / 16)   // 64 code tiles of 16
#define KSTEPS     (EMBED_DIM / 4)    // 16 WMMA k-steps (K=4 each)

// 32-bit LDS byte offset for async-to-LDS destination operand.
static __device__ __forceinline__ unsigned to_lds(const void* p) {
  return (unsigned)(unsigned long long)(__attribute__((address_space(3))) const void*)p;
}

static __device__ __forceinline__ void wait_async0() {
  asm volatile("s_wait_asynccnt 0x0" ::: "memory");
}

// ---------------------------------------------------------------------------
// c2[n*1024 + k] = sum_d codebook[n][k][d]^2
// ---------------------------------------------------------------------------
__global__ void c2_kernel(const float* __restrict__ cb, float* __restrict__ c2) {
  int e = blockIdx.x * blockDim.x + threadIdx.x;
  if (e >= NUM_CODES * BOOK_SIZE) return;
  const float4* p = (const float4*)(cb + (size_t)e * EMBED_DIM);
  float s = 0.f;
#pragma unroll
  for (int i = 0; i < EMBED_DIM / 4; ++i) {
    float4 v = p[i];
    s += v.x * v.x + v.y * v.y + v.z * v.z + v.w * v.w;
  }
  c2[e] = s;
}

// ---------------------------------------------------------------------------
// Main kernel: block = 128 threads (4 wave32), fixed code group n = blockIdx.y.
// Wave w handles rows [blockIdx.x*64 + w*16, +16). GEMM X(16x64) * C^T(64x16)
// per code tile via chained v_wmma_f32_16x16x4_f32, then running argmin.
// ---------------------------------------------------------------------------
__global__ __launch_bounds__(128) void vq_kernel(
    const float* __restrict__ x, const float* __restrict__ cb,
    const float* __restrict__ c2g, float* __restrict__ cw,
    float* __restrict__ onehot) {
  __shared__ __align__(16) float lds[2][16 * LDS_STRIDE];

  const int tid     = threadIdx.x;
  const int lane    = tid & 31;
  const int wave    = tid >> 5;
  const int n       = blockIdx.y;
  const int rowBase = blockIdx.x * 64 + wave * 16;
  const int colc    = lane & 15;   // A: row M=colc ; B/C: column N=colc
  const int half    = lane >> 4;   // K-pair select (0 -> K0,K1 ; 1 -> K2,K3)

  // ---- A fragments: x[rowBase+colc, n*64 + k], held in VGPRs for all 64 tiles.
  // ISA 16x4 f32 A layout: lane<16 holds {K=4kk+0, 4kk+1}, lane>=16 {4kk+2, 4kk+3}.
  v2f a[KSTEPS];
  {
    const float* xr = x + (size_t)(rowBase + colc) * (NUM_CODES * EMBED_DIM)
                        + n * EMBED_DIM + half * 2;
#pragma unroll
    for (int kk = 0; kk < KSTEPS; ++kk) {
      float2 t = *(const float2*)(xr + kk * 4);
      a[kk] = v2f{t.x, t.y};
    }
  }

  // Async stage of one 16-code B tile (4 KB) straight into LDS:
  // 128 threads x 2 x B128, no VGPR round-trip, tracked by ASYNCcnt.
  auto stage = [&](int t, int buf) {
    int q = tid;
#pragma unroll
    for (int it = 0; it < 2; ++it, q += 128) {
      int c  = q >> 4;
      int dq = q & 15;
      const float* g = cb +
          ((size_t)(n * BOOK_SIZE + t * 16 + c) * EMBED_DIM + dq * 4);
      unsigned l = to_lds(&lds[buf][c * LDS_STRIDE + dq * 4]);
      asm volatile("global_load_async_to_lds_b128 %0, %1, off"
                   :: "v"(l), "v"(g) : "memory");
    }
  };

  float minv[8];
  int   mini[8];
#pragma unroll
  for (int i = 0; i < 8; ++i) { minv[i] = 3.4e38f; mini[i] = 0; }

  stage(0, 0);
  wait_async0();
  __syncthreads();

  for (int t = 0; t < NTILES; ++t) {
    const int cur = t & 1;
    if (t + 1 < NTILES) stage(t + 1, cur ^ 1);   // async prefetch other buffer

    const float  c2v = c2g[n * BOOK_SIZE + t * 16 + colc];
    const float* bp  = &lds[cur][colc * LDS_STRIDE + half * 2];

    // Preload all B fragments so the 16 DS loads pipeline together and the
    // WMMA chains run without per-pair s_wait_dscnt stalls.
    v2f b[KSTEPS];
#pragma unroll
    for (int kk = 0; kk < KSTEPS; ++kk) {
      float2 bt = *(const float2*)(bp + kk * 4);
      b[kk] = v2f{bt.x, bt.y};
    }

    // Two independent accumulation chains (even/odd k-steps) for XDL ILP.
    v8f acc0 = {0.f, 0.f, 0.f, 0.f, 0.f, 0.f, 0.f, 0.f};
    v8f acc1 = {0.f, 0.f, 0.f, 0.f, 0.f, 0.f, 0.f, 0.f};
#pragma unroll
    for (int kk = 0; kk < KSTEPS; kk += 2) {
      acc0 = __builtin_amdgcn_wmma_f32_16x16x4_f32(
          false, a[kk], false, b[kk], (short)0, acc0, false, false);
      acc1 = __builtin_amdgcn_wmma_f32_16x16x4_f32(
          false, a[kk + 1], false, b[kk + 1], (short)0, acc1, false, false);
    }

    // dist(b,k) ~ c2[k] - 2*xc  (x^2 constant over k -> irrelevant for argmin)
    const int k = t * 16 + colc;
#pragma unroll
    for (int i = 0; i < 8; ++i) {
      float s = fmaf(-2.f, acc0[i] + acc1[i], c2v);
      if (s < minv[i]) { minv[i] = s; mini[i] = k; }   // ascending k => first-min
    }

    wait_async0();     // our prefetch landed in LDS
    __syncthreads();   // everyone's landed; safe to consume / later overwrite
  }

  // Cross-lane argmin over the 16 columns in each lane half (xor masks < 16
  // stay within a half). Ties -> smaller index, matching jnp.argmin.
#pragma unroll
  for (int m = 8; m >= 1; m >>= 1) {
#pragma unroll
    for (int i = 0; i < 8; ++i) {
      float ov = __shfl_xor(minv[i], m, 32);
      int   oi = __shfl_xor(mini[i], m, 32);
      if (ov < minv[i] || (ov == minv[i] && oi < mini[i])) {
        minv[i] = ov; mini[i] = oi;
      }
    }
  }

  // C layout: lanes 0-15 slot i -> row i ; lanes 16-31 slot i -> row i+8.
  // Broadcast each row's index, gather its codeword (64 floats, 2/lane), and
  // drop a 1.0 into the (memset-zeroed) one-hot.
#pragma unroll
  for (int r = 0; r < 16; ++r) {
    int idx = __shfl(mini[r & 7], (r < 8) ? 0 : 16, 32);
    int row = rowBase + r;
    float2 w = *(const float2*)(cb +
        ((size_t)(n * BOOK_SIZE + idx) * EMBED_DIM) + lane * 2);
    *(float2*)(cw + (size_t)row * (NUM_CODES * EMBED_DIM)
                   + n * EMBED_DIM + lane * 2) = w;
    if (lane == 0)
      onehot[((size_t)row * NUM_CODES + n) * BOOK_SIZE + idx] = 1.0f;
  }
}

// ---------------------------------------------------------------------------
extern "C" void kernel_launch(void* const* d_in, const int* in_sizes, int n_in,
                              void* d_out, int out_size, void* d_ws,
                              size_t ws_size, hipStream_t stream) {
  const float* x  = (const float*)d_in[0];   // (8192, 1024) f32
  const float* cb = (const float*)d_in[1];   // (16, 1024, 64) f32

  float* cw = (float*)d_out;                                   // (8192, 1024)
  float* oh = cw + (size_t)BATCH * NUM_CODES * EMBED_DIM;      // (8192,16,1024)
  float* c2 = (float*)d_ws;                                    // 16384 floats

  // One-hot is ~all zeros: bulk clear then scatter 128K ones in vq_kernel.
  hipMemsetAsync(oh, 0, (size_t)BATCH * NUM_CODES * BOOK_SIZE * sizeof(float),
                 stream);

  c2_kernel<<<(NUM_CODES * BOOK_SIZE + 255) / 256, 256, 0, stream>>>(cb, c2);

  dim3 grid(BATCH / 64, NUM_CODES);
  vq_kernel<<<grid, 128, 0, stream>>>(x, cb, c2, cw, oh);
}